// FConvLayer_56599079027314
// MI455X (gfx1250) — compile-verified
//
#include <hip/hip_runtime.h>

typedef float v2f __attribute__((ext_vector_type(2)));
typedef float v8f __attribute__((ext_vector_type(8)));

#define S_LEN 2048
#define HID   512
#define KSZ   64
#define HC    32      // channels per workgroup
#define TS    256     // sequence outputs per workgroup
#define WIN   320     // 319 real window entries + 1 guard (read under zero weight)
#define XROW  339     // swizzled x row stride (odd, >= sw(319)+1 = 339)
#define WROW  97      // padded weight row stride (odd)

__device__ __forceinline__ int sw(int i) { return i + (i >> 4); }

// Kernel 1: depthwise circular conv (via banded-Toeplitz WMMA) + residual.
// Writes h_res = conv*1/sqrt(S) + x into `out` (normalized in-place by kernel 2).
__global__ __launch_bounds__(256) void fconv_wmma_kernel(
    const float* __restrict__ x, const float* __restrict__ w,
    float* __restrict__ out)
{
  __shared__ float xs[HC][XROW];   // x window, swizzled columns
  __shared__ float ws[HC][WROW];   // weights at [16+k], zero pad elsewhere

  const int h0 = blockIdx.x * HC;
  const int s0 = blockIdx.y * TS;
  const int b  = blockIdx.z;
  const int t  = threadIdx.x;

  // zero the padded weight table
  for (int i = t; i < HC * WROW; i += 256) ((float*)ws)[i] = 0.0f;
  __syncthreads();

  {
    const int h  = t & (HC - 1);     // 0..31, consecutive threads -> consecutive h (coalesced)
    const int i0 = t >> 5;           // 0..7
    for (int i = i0; i < WIN; i += 8) {
      int sg = (s0 - 63 + i) & (S_LEN - 1);            // circular wrap
      xs[h][sw(i)] = x[((size_t)b * S_LEN + sg) * HID + h0 + h];
    }
    for (int k = i0; k < KSZ; k += 8) {
      ws[h][16 + k] = w[k * HID + h0 + h];             // conv_weight[0][k][h]
    }
  }
  __syncthreads();

  const int lane  = t & 31;
  const int wave  = t >> 5;
  const int nn    = lane & 15;       // M index for A-operand, N index for B-operand
  const int hi    = lane >> 4;       // 0: K sub 0/1, 1: K sub 2/3
  const int abase = 79 + nn - 2 * hi;                  // ws index = 16 + (m + 63 - c)
  const float scale = 0.022097086912079608f;           // 1/sqrt(2048) ('ortho' norm)

  for (int j = 0; j < HC / 8; ++j) {                   // 4 channels per wave
    const int h = wave * (HC / 8) + j;
    const float* __restrict__ wsr = ws[h];
    const float* __restrict__ xsr = xs[h];
    v8f acc = {0.f, 0.f, 0.f, 0.f, 0.f, 0.f, 0.f, 0.f};
#pragma unroll
    for (int tt = 0; tt < 20; ++tt) {                  // contraction K = 80 = 20 x 4
      v2f a, bb;
      a.x = wsr[abase - 4 * tt];                       // A[m, 4t + 2hi]
      a.y = wsr[abase - 4 * tt - 1];                   // A[m, 4t + 2hi + 1]
      int i0 = 16 * nn + 4 * tt + 2 * hi;              // B window index
      int i1 = i0 + 1;
      bb.x = xsr[i0 + (i0 >> 4)];
      bb.y = xsr[i1 + (i1 >> 4)];
      acc = __builtin_amdgcn_wmma_f32_16x16x4_f32(
          /*neg_a=*/false, a, /*neg_b=*/false, bb,
          /*c_mod=*/(short)0, acc, /*reuse_a=*/false, /*reuse_b=*/false);
    }
#pragma unroll
    for (int r = 0; r < 8; ++r) {                      // D: VGPR r -> M = r + 8*hi, N = nn
      int sl = (r + 8 * hi) + 16 * nn;                 // local s = m + 16n
      int iw = sl + 63;
      float xv = xsr[iw + (iw >> 4)];                  // residual from the same window
      out[((size_t)b * S_LEN + s0 + sl) * HID + h0 + h] = acc[r] * scale + xv;
    }
  }
}

// Kernel 2: in-place LayerNorm over the hidden dim (one block per (b,s) row).
__global__ __launch_bounds__(512) void ln_kernel(
    float* __restrict__ out,
    const float* __restrict__ gamma, const float* __restrict__ beta)
{
  const int s = blockIdx.x;
  const int b = blockIdx.y;
  const size_t base = ((size_t)b * S_LEN + s) * HID;
  const int h    = threadIdx.x;     // 0..511
  const int lane = h & 31;
  const int wave = h >> 5;

  __shared__ float2 part[16];
  __shared__ float2 mv;

  float v  = out[base + h];
  float sx = v, sq = v * v;
#pragma unroll
  for (int off = 16; off; off >>= 1) {
    sx += __shfl_xor(sx, off, 32);
    sq += __shfl_xor(sq, off, 32);
  }
  if (lane == 0) part[wave] = make_float2(sx, sq);
  __syncthreads();
  if (h == 0) {
    float a = 0.f, c = 0.f;
    for (int i = 0; i < 16; ++i) { a += part[i].x; c += part[i].y; }
    float mean = a * (1.0f / HID);
    float var  = c * (1.0f / HID) - mean * mean;
    mv = make_float2(mean, rsqrtf(var + 1e-12f));
  }
  __syncthreads();
  out[base + h] = gamma[h] * (v - mv.x) * mv.y + beta[h];
}

extern "C" void kernel_launch(void* const* d_in, const int* in_sizes, int n_in,
                              void* d_out, int out_size, void* d_ws, size_t ws_size,
                              hipStream_t stream) {
  (void)in_sizes; (void)n_in; (void)out_size; (void)d_ws; (void)ws_size;
  const float* x  = (const float*)d_in[0];   // [32, 2048, 512] f32
  const float* w  = (const float*)d_in[1];   // [1, 64, 512]   f32
  const float* g  = (const float*)d_in[2];   // [512]          f32
  const float* be = (const float*)d_in[3];   // [512]          f32
  float* out = (float*)d_out;                // [32, 2048, 512] f32

  dim3 grid1(HID / HC, S_LEN / TS, 32);      // (16, 8, 32) workgroups
  fconv_wmma_kernel<<<grid1, 256, 0, stream>>>(x, w, out);

  dim3 grid2(S_LEN, 32);                     // one row per block
  ln_kernel<<<grid2, HID, 0, stream>>>(out, g, be);
}